// MambaWrapper_11811160064390
// MI455X (gfx1250) — compile-verified
//
#include <hip/hip_runtime.h>
#include <hip/hip_bf16.h>
#include <math.h>

// ---------------- problem constants ----------------
#define BATCH   32
#define TLEN    4096
#define BT      (BATCH * TLEN)      // 131072 tokens
#define DMODEL  128
#define OBSD    256
#define DINNER  256
#define DSTATE  64
#define NHEADS  4
#define HEADDIM 64
#define DCONVDIM 384                // DINNER + 2*DSTATE
#define NPROJ   644                 // 2*DINNER + 2*DSTATE + NHEADS
#define ZXB_LD  640                 // stored width of zxbcdt (z[256] + xBC[384])
#define NACT    64

typedef __attribute__((ext_vector_type(2))) float v2f;
typedef __attribute__((ext_vector_type(8))) float v8f;

// ---------------- CDNA5 async copy to LDS (ASYNCcnt-tracked) ----------------
__device__ __forceinline__ unsigned lds_off(const void* p) {
  // generic LDS pointer = {shared aperture, lds offset}; low 32 bits = offset
  return (unsigned)(uintptr_t)p;
}
__device__ __forceinline__ void async_ld_b128(unsigned lds, const void* g) {
  asm volatile("global_load_async_to_lds_b128 %0, %1, off"
               :: "v"(lds), "v"(g) : "memory");
}
__device__ __forceinline__ void async_ld_b32(unsigned lds, const void* g) {
  asm volatile("global_load_async_to_lds_b32 %0, %1, off"
               :: "v"(lds), "v"(g) : "memory");
}
__device__ __forceinline__ void wait_async0() {
  asm volatile("s_wait_asynccnt 0x0" ::: "memory");
}

// ---------------------------------------------------------------------------
// K0: collapse encoder into in-proj:  Wc[256,644] = W_enc @ W_in,
//     bc[644] = b_enc @ W_in   (row k==OBSD computes bc)
// ---------------------------------------------------------------------------
__global__ void prep_wc_kernel(const float* __restrict__ W_enc,
                               const float* __restrict__ W_in,
                               const float* __restrict__ b_enc,
                               float* __restrict__ Wc,
                               float* __restrict__ bc) {
  int idx = blockIdx.x * blockDim.x + threadIdx.x;
  const int total = (OBSD + 1) * NPROJ;
  if (idx >= total) return;
  int k = idx / NPROJ;
  int n = idx % NPROJ;
  float acc = 0.f;
  if (k < OBSD) {
    #pragma unroll 8
    for (int m = 0; m < DMODEL; ++m)
      acc += W_enc[k * DMODEL + m] * W_in[m * NPROJ + n];
    Wc[k * NPROJ + n] = acc;
  } else {
    #pragma unroll 8
    for (int m = 0; m < DMODEL; ++m)
      acc += b_enc[m] * W_in[m * NPROJ + n];
    bc[n] = acc;
  }
}

// ---------------------------------------------------------------------------
// fp32 WMMA GEMM with LDS-staged weight panel.
// Block = 256 threads = 8 waves; block computes 256 rows x 64 cols.
// The K x 64 weight strip is staged once into LDS, packed so the (k,k+1)
// fragment pair a lane needs is one ds_load_b64. Each wave runs TWO 16-row
// tiles against each B fragment (8 v_wmma per A load pair).
// ---------------------------------------------------------------------------
__global__ void __launch_bounds__(256)
wmma_gemm_f32_kernel(const float* __restrict__ A, int lda,
                     const float* __restrict__ Bw, int ldb,
                     float* __restrict__ C, int ldc,
                     int K, int nstrips,
                     const float* __restrict__ bias) {
  extern __shared__ float2 sB2[];            // (K/2) * 64 float2
  const int tid     = threadIdx.x;
  const int strip   = blockIdx.x % nstrips;
  const int rowGrp  = blockIdx.x / nstrips;
  const int colbase = strip << 6;

  // stage weight strip: element (k, col) -> pair (k>>1)*64+col, slot k&1
  for (int i = tid; i < K * 64; i += 256) {
    int k = i >> 6, col = i & 63;
    ((float*)sB2)[(((k >> 1) * 64) + col) * 2 + (k & 1)] =
        Bw[(size_t)k * ldb + colbase + col];
  }
  __syncthreads();

  const int wave = tid >> 5, lane = tid & 31;
  const int mn = lane & 15;                  // row / col within tile
  const int kh = lane >> 4;                  // which K pair this half holds
  const int rowbase = (rowGrp * 16 + wave * 2) * 16;

  const float* Arow0 = A + (size_t)(rowbase + mn) * lda;
  const float* Arow1 = Arow0 + (size_t)16 * lda;

  v8f acc[2][4] = {};

  #pragma unroll 4
  for (int k0 = 0; k0 < K; k0 += 4) {
    const int ka = k0 + 2 * kh;              // even -> 8B-aligned v2f loads
    v2f a0 = *(const v2f*)(Arow0 + ka);
    v2f a1 = *(const v2f*)(Arow1 + ka);
    const float2* bp = sB2 + ((k0 >> 1) + kh) * 64 + mn;
    #pragma unroll
    for (int t = 0; t < 4; ++t) {
      float2 bb = bp[16 * t];                // ds_load_b64
      v2f b; b.x = bb.x; b.y = bb.y;
      acc[0][t] = __builtin_amdgcn_wmma_f32_16x16x4_f32(
          false, a0, false, b, (short)0, acc[0][t], false, false);
      acc[1][t] = __builtin_amdgcn_wmma_f32_16x16x4_f32(
          false, a1, false, b, (short)0, acc[1][t], false, false);
    }
  }

  #pragma unroll
  for (int t = 0; t < 4; ++t) {
    const int col = colbase + 16 * t + mn;
    const float bv = bias ? bias[col] : 0.f;
    #pragma unroll
    for (int m = 0; m < 2; ++m) {
      float* Cp = C + (size_t)(rowbase + m * 16 + 8 * kh) * ldc + col;
      #pragma unroll
      for (int r = 0; r < 8; ++r)
        Cp[(size_t)r * ldc] = acc[m][t][r] + bv;
    }
  }
}

// ---------------------------------------------------------------------------
// K2: dt / dA per (token, head):  raw = obs_row . Wc[:,640+h] + bc[640+h]
// ---------------------------------------------------------------------------
__global__ void dtda_kernel(const float* __restrict__ obs,
                            const float* __restrict__ Wc,
                            const float* __restrict__ bc,
                            const float* __restrict__ dt_bias,
                            const float* __restrict__ A_log,
                            float* __restrict__ dtb,
                            float* __restrict__ dAb) {
  int idx = blockIdx.x * blockDim.x + threadIdx.x;   // BT * NHEADS
  if (idx >= BT * NHEADS) return;
  int h = idx & (NHEADS - 1);
  size_t row = (size_t)(idx >> 2);
  const float* o = obs + row * OBSD;
  float acc = bc[ZXB_LD + h];
  #pragma unroll 8
  for (int k = 0; k < OBSD; ++k)
    acc += o[k] * Wc[(size_t)k * NPROJ + ZXB_LD + h];
  float v  = acc + dt_bias[h];
  float dt = (v > 20.f) ? v : log1pf(__expf(v));
  float A  = -__expf(A_log[h]);
  dtb[idx] = dt;
  dAb[idx] = __expf(dt * A);
}

// ---------------------------------------------------------------------------
// K3: depthwise causal conv(4) over time on xBC + bias + SiLU; split x/B/C.
// ---------------------------------------------------------------------------
__global__ void conv_silu_kernel(const float* __restrict__ zxb,
                                 const float* __restrict__ W_conv,
                                 const float* __restrict__ b_conv,
                                 float* __restrict__ xb,
                                 float* __restrict__ Bb,
                                 float* __restrict__ Cb) {
  long long idx = (long long)blockIdx.x * blockDim.x + threadIdx.x;
  if (idx >= (long long)BT * DCONVDIM) return;
  int    c   = (int)(idx % DCONVDIM);
  size_t row = (size_t)(idx / DCONVDIM);   // b*T + t
  int    t   = (int)(row & (TLEN - 1));
  float acc = b_conv[c];
  #pragma unroll
  for (int j = 0; j < 4; ++j) {
    int tt = t - 3 + j;
    if (tt >= 0)
      acc += zxb[(row - 3 + (size_t)j) * ZXB_LD + DINNER + c] * W_conv[c * 4 + j];
  }
  float s = acc / (1.f + __expf(-acc));    // SiLU
  if (c < DINNER)                 xb[row * DINNER + c] = s;
  else if (c < DINNER + DSTATE)   Bb[row * DSTATE + (c - DINNER)] = s;
  else                            Cb[row * DSTATE + (c - DINNER - DSTATE)] = s;
}

// ---------------------------------------------------------------------------
// K4: selective scan. One block per (batch, head); 256 threads; thread holds
// 16 state elements s[p, n0..n0+15] in VGPRs. Chunks of CT timesteps are
// DOUBLE-BUFFERED in LDS, filled by global_load_async_to_lds_b128 (ASYNCcnt)
// so the next chunk's traffic overlaps the serial recurrence.
// y (+ D*x) written IN PLACE over x (this block owns its 64-col slice).
// ---------------------------------------------------------------------------
#define CT 32
#define CHUNK_F (CT * HEADDIM)    // 2048 floats per staged array

__global__ void __launch_bounds__(256)
scan_kernel(float* __restrict__ xyb,          // [BT,256] x in, y out (in place)
            const float* __restrict__ Bb,     // [BT,64]
            const float* __restrict__ Cb,     // [BT,64]
            const float* __restrict__ dtb,    // [BT,4]
            const float* __restrict__ dAb,    // [BT,4]
            const float* __restrict__ Dvec) { // [4]
  __shared__ float sx[2][CHUNK_F];
  __shared__ float sB[2][CHUNK_F];
  __shared__ float sC[2][CHUNK_F];
  __shared__ float sdt[2][CT];
  __shared__ float sdA[2][CT];

  const int bh  = blockIdx.x;
  const int b   = bh >> 2;
  const int h   = bh & 3;
  const int tid = threadIdx.x;
  const int ng  = tid & 3;          // which 16-wide n group
  const int p   = tid >> 2;         // head dim 0..63
  const int n0  = ng << 4;
  const size_t base = (size_t)b * TLEN;

  float s[16];
  #pragma unroll
  for (int j = 0; j < 16; ++j) s[j] = 0.f;
  const float Dh = Dvec[h];

  auto issue = [&](int buf, int t0) {
    #pragma unroll
    for (int r = 0; r < 2; ++r) {
      int f = tid + r * 256;        // float4 index, 512 per array
      int e = f * 4;
      int tt = e >> 6, pp = e & 63;
      size_t row = base + t0 + tt;
      async_ld_b128(lds_off(&sx[buf][e]), xyb + row * DINNER + h * HEADDIM + pp);
      async_ld_b128(lds_off(&sB[buf][e]), Bb + row * DSTATE + pp);
      async_ld_b128(lds_off(&sC[buf][e]), Cb + row * DSTATE + pp);
    }
    if (tid < CT) {
      size_t row = base + t0 + tid;
      async_ld_b32(lds_off(&sdt[buf][tid]), dtb + row * NHEADS + h);
      async_ld_b32(lds_off(&sdA[buf][tid]), dAb + row * NHEADS + h);
    }
  };

  issue(0, 0);
  int cur = 0;
  for (int t0 = 0; t0 < TLEN; t0 += CT) {
    wait_async0();                  // my chunk's async fills done
    __syncthreads();                // everyone's fills done, prev reads done
    if (t0 + CT < TLEN) issue(cur ^ 1, t0 + CT);

    for (int tt = 0; tt < CT; ++tt) {
      const float dA  = sdA[cur][tt];
      const float dtv = sdt[cur][tt];
      const float xp  = sx[cur][tt * HEADDIM + p];
      const float dtx = dtv * xp;
      float yp = 0.f;
      const float4* B4 = (const float4*)(&sB[cur][tt * DSTATE + n0]);
      const float4* C4 = (const float4*)(&sC[cur][tt * DSTATE + n0]);
      #pragma unroll
      for (int q = 0; q < 4; ++q) {
        float4 Bv = B4[q], Cv = C4[q];
        s[4*q+0] = fmaf(dA, s[4*q+0], dtx * Bv.x); yp = fmaf(s[4*q+0], Cv.x, yp);
        s[4*q+1] = fmaf(dA, s[4*q+1], dtx * Bv.y); yp = fmaf(s[4*q+1], Cv.y, yp);
        s[4*q+2] = fmaf(dA, s[4*q+2], dtx * Bv.z); yp = fmaf(s[4*q+2], Cv.z, yp);
        s[4*q+3] = fmaf(dA, s[4*q+3], dtx * Bv.w); yp = fmaf(s[4*q+3], Cv.w, yp);
      }
      // reduce over the 4 lanes sharing p (wave32 shuffles)
      yp += __shfl_xor(yp, 1, 32);
      yp += __shfl_xor(yp, 2, 32);
      if (ng == 0)
        xyb[(base + t0 + tt) * DINNER + h * HEADDIM + p] = yp + Dh * xp;
    }
    cur ^= 1;
  }
}

// ---------------------------------------------------------------------------
// K5: g = y * silu(z); RMS-norm over 256; write g IN PLACE over z.
// ---------------------------------------------------------------------------
__global__ void gate_norm_kernel(const float* __restrict__ yb,   // [BT,256]
                                 float* __restrict__ zxb,        // [BT,ZXB_LD]
                                 const float* __restrict__ norm_w) {
  const int w    = (blockIdx.x * blockDim.x + threadIdx.x) >> 5;
  const int lane = threadIdx.x & 31;
  const size_t row = (size_t)w;
  const float* yr = yb + row * DINNER;
  float* zr = zxb + row * ZXB_LD;
  float gv[8];
  float ss = 0.f;
  #pragma unroll
  for (int i = 0; i < 8; ++i) {
    int c = lane + i * 32;
    float zz = zr[c];
    float v  = yr[c] * (zz / (1.f + __expf(-zz)));
    gv[i] = v;
    ss += v * v;
  }
  #pragma unroll
  for (int off = 16; off; off >>= 1) ss += __shfl_xor(ss, off, 32);
  const float scale = rsqrtf(ss * (1.f / 256.f) + 1e-5f);
  #pragma unroll
  for (int i = 0; i < 8; ++i) {
    int c = lane + i * 32;
    zr[c] = gv[i] * scale * norm_w[c];   // same locations already read above
  }
}

// ---------------------------------------------------------------------------
// K8: values = flat @ W_val + b_val
// ---------------------------------------------------------------------------
__global__ void value_kernel(const float* __restrict__ flat,
                             const float* __restrict__ W_val,
                             const float* __restrict__ b_val,
                             float* __restrict__ outv) {
  int row = blockIdx.x * blockDim.x + threadIdx.x;
  if (row >= BT) return;
  const float* f = flat + (size_t)row * DMODEL;
  float acc = b_val[0];
  #pragma unroll 8
  for (int k = 0; k < DMODEL; ++k) acc += f[k] * W_val[k];
  outv[row] = acc;
}

// ---------------------------------------------------------------------------
extern "C" void kernel_launch(void* const* d_in, const int* in_sizes, int n_in,
                              void* d_out, int out_size, void* d_ws, size_t ws_size,
                              hipStream_t stream) {
  const float* obs     = (const float*)d_in[0];
  const float* W_enc   = (const float*)d_in[1];
  const float* b_enc   = (const float*)d_in[2];
  const float* W_in    = (const float*)d_in[3];
  const float* W_conv  = (const float*)d_in[4];
  const float* b_conv  = (const float*)d_in[5];
  const float* dt_bias = (const float*)d_in[6];
  const float* A_log   = (const float*)d_in[7];
  const float* Dvec    = (const float*)d_in[8];
  const float* norm_w  = (const float*)d_in[9];
  const float* W_out   = (const float*)d_in[10];
  const float* W_act   = (const float*)d_in[11];
  const float* b_act   = (const float*)d_in[12];
  const float* W_val   = (const float*)d_in[13];
  const float* b_val   = (const float*)d_in[14];
  float* out = (float*)d_out;

  // workspace layout (fp32), peak ~541 MB
  float* ws  = (float*)d_ws;
  float* zxb = ws;                               // [BT,640]  z + xBC (g overwrites z)
  float* xb  = zxb + (size_t)BT * ZXB_LD;        // [BT,256]  x -> y -> flat
  float* Bb  = xb  + (size_t)BT * DINNER;        // [BT,64]
  float* Cb  = Bb  + (size_t)BT * DSTATE;        // [BT,64]
  float* dtb = Cb  + (size_t)BT * DSTATE;        // [BT,4]
  float* dAb = dtb + (size_t)BT * NHEADS;        // [BT,4]
  float* Wc  = dAb + (size_t)BT * NHEADS;        // [256,644]
  float* bc  = Wc  + (size_t)OBSD * NPROJ;       // [644]

  // K0: collapsed encoder+in-proj weights
  {
    int total = (OBSD + 1) * NPROJ;
    prep_wc_kernel<<<(total + 255) / 256, 256, 0, stream>>>(W_enc, W_in, b_enc, Wc, bc);
  }
  // K1: zxbcdt[:, 0:640] = obs @ Wc + bc   (WMMA, 10 strips of 64 cols)
  {
    int nstrips = ZXB_LD / 64;                   // 10
    int blocks  = (BT / 256) * nstrips;          // 5120
    wmma_gemm_f32_kernel<<<blocks, 256, OBSD * 64 * sizeof(float), stream>>>(
        obs, OBSD, Wc, NPROJ, zxb, ZXB_LD, OBSD, nstrips, bc);
  }
  // K2: dt / dA (last 4 in-proj columns fused here)
  dtda_kernel<<<(BT * NHEADS + 255) / 256, 256, 0, stream>>>(
      obs, Wc, bc, dt_bias, A_log, dtb, dAb);
  // K3: causal conv + SiLU + split
  {
    long long total = (long long)BT * DCONVDIM;
    conv_silu_kernel<<<(unsigned)((total + 255) / 256), 256, 0, stream>>>(
        zxb, W_conv, b_conv, xb, Bb, Cb);
  }
  // K4: selective scan (async-LDS double buffered), y (+D*x) overwrites x
  scan_kernel<<<BATCH * NHEADS, 256, 0, stream>>>(xb, Bb, Cb, dtb, dAb, Dvec);
  // K5: gate + RMS norm, g overwrites z inside zxb
  gate_norm_kernel<<<BT / 8, 256, 0, stream>>>(xb, zxb, norm_w);
  // K6: flat = g @ W_out   (WMMA, 2 strips) -> reuse xb
  {
    int blocks = (BT / 256) * 2;                 // 1024
    wmma_gemm_f32_kernel<<<blocks, 256, DINNER * 64 * sizeof(float), stream>>>(
        zxb, ZXB_LD, W_out, DMODEL, xb, DMODEL, DINNER, 2, nullptr);
  }
  // K7: logits = flat @ W_act + b_act   (WMMA, 1 strip) -> d_out
  {
    int blocks = (BT / 256) * 1;                 // 512
    wmma_gemm_f32_kernel<<<blocks, 256, DMODEL * 64 * sizeof(float), stream>>>(
        xb, DMODEL, W_act, NACT, out, NACT, DMODEL, 1, b_act);
  }
  // K8: values -> d_out tail
  value_kernel<<<BT / 256, 256, 0, stream>>>(xb, W_val, b_val, out + (size_t)BT * NACT);
}